// StochasticAssociativeCuBaLIF_17540646436866
// MI455X (gfx1250) — compile-verified
//
#include <hip/hip_runtime.h>
#include <stdint.h>

// Problem shape (fixed by the reference): u[B=16, T=1024, H=2048] f32.
#define T_STEPS 1024
#define H_DIM   2048
#define B_DIM   16
#define ROWS    8                      // timesteps per chunk
#define DEPTH   7                      // chunks in flight (56 async loads <= 63 cap)
#define NCHUNK  (T_STEPS / ROWS)       // 128
#define WPB     2                      // waves per block (independent, no barriers)
#define BLOCK   (WPB * 32)

// --- CDNA5 async global->LDS copy (gfx1250), tracked by ASYNCcnt ---------
__device__ __forceinline__ void async_load_f32(uint32_t lds_addr, uint64_t gaddr) {
    asm volatile("global_load_async_to_lds_b32 %0, %1, off"
                 :: "v"(lds_addr), "v"(gaddr)
                 : "memory");
}
template <int N>
__device__ __forceinline__ void wait_async_le() {
    asm volatile("s_wait_asynccnt %0" :: "n"(N) : "memory");
}
__device__ __forceinline__ void wait_ds_0() {
    asm volatile("s_wait_dscnt 0" ::: "memory");
}

struct ScanState {
    float x, V;
    uint32_t idx;
};

// Consume one 8-row chunk from LDS: scan + spike + store.
__device__ __forceinline__ void compute_chunk(const float (*cbuf)[32], int lane,
                                              float a, float bb, uint32_t seed,
                                              float* ob, ScanState& st)
{
#pragma unroll
    for (int r = 0; r < ROWS; ++r) {
        const float uu = cbuf[r][lane];              // ds_load (pairs into 2addr)
        st.x = fmaf(a,  st.x, uu);                   // synaptic current scan
        st.V = fmaf(bb, st.V, st.x);                 // membrane potential scan
        const float U = st.V - 1.0f;                 // THRESHOLD = 1.0
        // v_cos_f32 computes cos(2*pi*src) == cos(FREQ*U)
        const float cc = __builtin_amdgcn_cosf(U);
        // sigmoid(100U) = 1/(1 + exp2(-100*log2e*U))
        const float e  = __builtin_amdgcn_exp2f(-144.2695040889f * U);
        const float s  = __builtin_amdgcn_rcpf(1.0f + e);
        const float p  = fmaxf(cc * s, 0.0f);
        // deterministic counter-based uniform in [0,1)
        uint32_t z = st.idx ^ (seed * 0x9E3779B9u);
        z = (z ^ (z >> 16)) * 0x7feb352du;
        z = (z ^ (z >> 15)) * 0x846ca68bu;
        z ^= z >> 16;
        const float rnd = (float)(z >> 8) * (1.0f / 16777216.0f);
        __builtin_nontemporal_store((rnd < p) ? 1.0f : 0.0f,
                                    ob + (size_t)r * H_DIM);
        st.idx += H_DIM;
    }
}

__global__ __launch_bounds__(BLOCK)
void snn_cuba_lif_kernel(const float* __restrict__ u,
                         const float* __restrict__ alpha,
                         const float* __restrict__ beta,
                         const int*   __restrict__ seed_p,
                         float*       __restrict__ out)
{
    // Per-wave ring of DEPTH chunk slots: [wave][slot][row(t)][lane(h)]
    __shared__ float buf[WPB][DEPTH][ROWS][32];

    const int tid  = threadIdx.x;
    const int wave = tid >> 5;
    const int lane = tid & 31;
    const int g    = blockIdx.x * BLOCK + tid;       // 0..32767 = one (b,h) sequence
    const int b    = g >> 11;                         // H=2048 => 11 bits
    const int h    = g & (H_DIM - 1);

    const uint32_t seed = (uint32_t)seed_p[0];

    float a  = alpha[h]; a  = fminf(fmaxf(a,  0.0f), 1.0f);
    float bb = beta[h];  bb = fminf(fmaxf(bb, 0.0f), 1.0f);

    const size_t seq_base = ((size_t)b * T_STEPS) * H_DIM + (size_t)h;
    const float* up = u   + seq_base;    // stride H_DIM per timestep
    float*       op = out + seq_base;

    // Low 32 bits of a generic pointer to __shared__ == LDS byte address.
    uint32_t lds_slot[DEPTH];
#pragma unroll
    for (int sl = 0; sl < DEPTH; ++sl)
        lds_slot[sl] = (uint32_t)(size_t)&buf[wave][sl][0][lane];

    // Prologue: fill the pipeline with chunks 0..DEPTH-1 (56 loads in flight).
#pragma unroll
    for (int c = 0; c < DEPTH; ++c) {
        const float* cp = up + (size_t)(c * ROWS) * H_DIM;
#pragma unroll
        for (int r = 0; r < ROWS; ++r)
            async_load_f32(lds_slot[c] + (uint32_t)(r * 128),
                           (uint64_t)(uintptr_t)(cp + (size_t)r * H_DIM));
    }

    ScanState st;
    st.x = 0.0f; st.V = 0.0f;
    st.idx = (((uint32_t)b * T_STEPS) << 11) + (uint32_t)h;  // linear elem index

    // Steady state: chunk c lands when <= (DEPTH-1)*ROWS loads remain (in-order).
    int slot = 0;
    for (int c = 0; c < NCHUNK - DEPTH; ++c) {
        wait_async_le<(DEPTH - 1) * ROWS>();
        compute_chunk(buf[wave][slot], lane, a, bb, seed,
                      op + (size_t)(c * ROWS) * H_DIM, st);
        wait_ds_0();   // LDS reads done before the async engine reuses this slot
        const float* np = up + (size_t)((c + DEPTH) * ROWS) * H_DIM;
#pragma unroll
        for (int r = 0; r < ROWS; ++r)
            async_load_f32(lds_slot[slot] + (uint32_t)(r * 128),
                           (uint64_t)(uintptr_t)(np + (size_t)r * H_DIM));
        slot = (slot + 1 == DEPTH) ? 0 : slot + 1;
    }

    // Drain: no more issues; step the wait threshold down 48 -> 0.
    // Wait immediates must be compile-time, so unroll explicitly.
#define DRAIN_STEP(K)                                                          \
    {                                                                          \
        wait_async_le<(DEPTH - 1 - (K)) * ROWS>();                             \
        compute_chunk(buf[wave][slot], lane, a, bb, seed,                      \
                      op + (size_t)((NCHUNK - DEPTH + (K)) * ROWS) * H_DIM,    \
                      st);                                                     \
        slot = (slot + 1 == DEPTH) ? 0 : slot + 1;                             \
    }
    DRAIN_STEP(0)
    DRAIN_STEP(1)
    DRAIN_STEP(2)
    DRAIN_STEP(3)
    DRAIN_STEP(4)
    DRAIN_STEP(5)
    DRAIN_STEP(6)
#undef DRAIN_STEP
}

extern "C" void kernel_launch(void* const* d_in, const int* in_sizes, int n_in,
                              void* d_out, int out_size, void* d_ws, size_t ws_size,
                              hipStream_t stream) {
    (void)in_sizes; (void)n_in; (void)out_size; (void)d_ws; (void)ws_size;
    const float* u     = (const float*)d_in[0];
    const float* alpha = (const float*)d_in[1];
    const float* beta  = (const float*)d_in[2];
    const int*   seed  = (const int*)d_in[3];
    float*       out   = (float*)d_out;

    const int total_seqs = B_DIM * H_DIM;           // 32768 sequences
    dim3 grid(total_seqs / BLOCK);                   // 512 blocks
    dim3 block(BLOCK);                               // 64 threads = 2 waves
    hipLaunchKernelGGL(snn_cuba_lif_kernel, grid, block, 0, stream,
                       u, alpha, beta, seed, out);
}